// ActionPredictionModel_83940840833283
// MI455X (gfx1250) — compile-verified
//
#include <hip/hip_runtime.h>
#include <hip/hip_bf16.h>
#include <math.h>

// ---------------------------------------------------------------------------
// MI455X (gfx1250, wave32) implementation.
// - All GEMMs on V_WMMA_F32_16X16X4_F32 (exact f32; problem is HBM-bound on
//   f32 weights, so low-precision WMMA buys nothing).
// - Dominant GEMM (specs @ W1, 6.5 MB of weights) double-buffers its W1 tiles
//   into LDS with GLOBAL_LOAD_ASYNC_TO_LDS_B128 (ASYNCcnt), overlapping the
//   DMA with WMMA compute.
// ---------------------------------------------------------------------------

typedef __attribute__((ext_vector_type(2))) float v2f;
typedef __attribute__((ext_vector_type(8))) float v8f;

static __device__ __forceinline__ v8f wmma4(v2f a, v2f b, v8f c) {
  // 8 args: (neg_a, A, neg_b, B, c_mod, C, reuse_a, reuse_b)
  return __builtin_amdgcn_wmma_f32_16x16x4_f32(false, a, false, b, (short)0, c,
                                               false, false);
}

// LDS byte offset of a generic shared pointer (low 32 bits of the address)
static __device__ __forceinline__ uint32_t lds_off(const void* p) {
  return (uint32_t)(uintptr_t)p;
}

// Async DMA: 16 bytes memory -> LDS, tracked by ASYNCcnt.
static __device__ __forceinline__ void async_load_b128(uint32_t ldsoff,
                                                       const void* gptr) {
  asm volatile("global_load_async_to_lds_b128 %0, %1, off"
               :
               : "v"(ldsoff), "v"(gptr)
               : "memory");
}

static __device__ __forceinline__ void wait_async0() {
  asm volatile("s_wait_asynccnt 0" ::: "memory");
}

// ============================ Kernel 1 =====================================
// s1[64,900] = relu(specs[64,1801] @ W1[1801,900] + b1)
// Block: 256 thr (8 waves) -> wave grid 4(M16) x 2(N32); block tile 64x64.
// K-chunks of 32, double-buffered: W1 tile via async DMA, specs tile regular.
__global__ void k_spec_gemm1(const float* __restrict__ specs,
                             const float* __restrict__ W1,
                             const float* __restrict__ b1,
                             float* __restrict__ s1) {
  __shared__ float lA[2][64][33];   // 64 rows x 32 k   (pad: conflict-free)
  __shared__ float lB[2][32][72];   // 32 k   x 64 n    (row 288B: 16B-aligned,
                                    //  2-row bank shift = 16 -> disjoint halves)
  const int tid = threadIdx.x;
  const int nbase_g = blockIdx.x * 64;            // 15 tiles cover 900
  const int w = tid >> 5, lane = tid & 31;
  const int lane15 = lane & 15, hi = lane >> 4;
  const int Mb = (w & 3) * 16, Nb = (w >> 2) * 32;
  v8f c0 = {}, c1 = {};

  auto stage = [&](int buf, int k0) {
    // --- B tile: W1[k0..k0+31][nbase..+63] via async B128 (16B segments) ---
    for (int e = tid; e < 32 * 16; e += 256) {
      int kk = e >> 4, seg = e & 15;
      int k = k0 + kk, nseg = nbase_g + seg * 4;
      float* dst = &lB[buf][kk][seg * 4];
      if (k < 1801 && nseg + 4 <= 900) {
        async_load_b128(lds_off(dst), W1 + (size_t)k * 900 + nseg);
      } else {
        dst[0] = 0.0f; dst[1] = 0.0f; dst[2] = 0.0f; dst[3] = 0.0f;
      }
    }
    // --- A tile: specs[64][k0..k0+31] (odd row stride 1801 -> regular) ---
    for (int e = tid; e < 64 * 32; e += 256) {
      int r = e >> 5, cc = e & 31, k = k0 + cc;
      lA[buf][r][cc] = (k < 1801) ? specs[r * 1801 + k] : 0.0f;
    }
  };

  stage(0, 0);
  wait_async0();
  __syncthreads();
  int buf = 0;
  for (int k0 = 0; k0 < 1801; k0 += 32) {
    if (k0 + 32 < 1801) stage(buf ^ 1, k0 + 32);   // overlap DMA with compute
#pragma unroll
    for (int kk = 0; kk < 32; kk += 4) {
      v2f a, b0, b1v;
      a.x  = lA[buf][Mb + lane15][kk + 2 * hi];
      a.y  = lA[buf][Mb + lane15][kk + 2 * hi + 1];
      b0.x = lB[buf][kk + 2 * hi][Nb + lane15];
      b0.y = lB[buf][kk + 2 * hi + 1][Nb + lane15];
      b1v.x = lB[buf][kk + 2 * hi][Nb + 16 + lane15];
      b1v.y = lB[buf][kk + 2 * hi + 1][Nb + 16 + lane15];
      c0 = wmma4(a, b0, c0);
      c1 = wmma4(a, b1v, c1);
    }
    wait_async0();
    __syncthreads();
    buf ^= 1;
  }
  for (int r = 0; r < 8; ++r) {
    int m = Mb + r + 8 * hi;
    int n0 = nbase_g + Nb + lane15, n1 = n0 + 16;
    if (n0 < 900) { float v = c0[r] + b1[n0]; s1[m * 900 + n0] = v > 0 ? v : 0; }
    if (n1 < 900) { float v = c1[r] + b1[n1]; s1[m * 900 + n1] = v > 0 ? v : 0; }
  }
}

// ============================ Kernel 2 =====================================
// s[64,100] = relu(s1[64,900] @ W2[900,100] + b2)   (single block)
__global__ void k_spec_gemm2(const float* __restrict__ s1,
                             const float* __restrict__ W2,
                             const float* __restrict__ b2,
                             float* __restrict__ s) {
  const int tid = threadIdx.x;
  const int w = tid >> 5, lane = tid & 31;
  const int lane15 = lane & 15, hi = lane >> 4;
  const int Mb = (w & 3) * 16, Nb = (w >> 2) * 64;
  v8f c[4] = {};
  for (int k = 0; k < 900; k += 4) {               // 900 % 4 == 0
    int ka = k + 2 * hi;
    v2f a;
    a.x = s1[(Mb + lane15) * 900 + ka];
    a.y = s1[(Mb + lane15) * 900 + ka + 1];
#pragma unroll
    for (int t = 0; t < 4; ++t) {
      int n = Nb + 16 * t + lane15;
      v2f b;
      b.x = (n < 100) ? W2[ka * 100 + n] : 0.0f;
      b.y = (n < 100) ? W2[(ka + 1) * 100 + n] : 0.0f;
      c[t] = wmma4(a, b, c[t]);
    }
  }
  for (int t = 0; t < 4; ++t)
    for (int r = 0; r < 8; ++r) {
      int m = Mb + r + 8 * hi, n = Nb + 16 * t + lane15;
      if (n < 100) { float v = c[t][r] + b2[n]; s[m * 100 + n] = v > 0 ? v : 0; }
    }
}

// ============================ Kernel 3 =====================================
// Ai = relu(nf) @ Wa2[0:128], Bj = relu(nf) @ Wa2[128:256]; grid (9, 2)
// Wa2 tile staged with async DMA (rows are 512B aligned).
__global__ void k_pair_proj(const float* __restrict__ nf,
                            const float* __restrict__ Wa2,
                            float* __restrict__ Ai, float* __restrict__ Bj) {
  __shared__ float lA[64][65];
  __shared__ float lB[64][136];    // row 544B: 16B-aligned; 2-row shift = 16 banks
  const int tid = threadIdx.x;
  const int mtile = blockIdx.x;    // 9 tiles of 64 rows == 576
  const int osel = blockIdx.y;     // 0 -> Ai, 1 -> Bj
  float* out = osel ? Bj : Ai;
  const int w = tid >> 5, lane = tid & 31;
  const int lane15 = lane & 15, hi = lane >> 4;
  const int Mb = (w & 3) * 16, Nb = (w >> 2) * 64;
  v8f c[4] = {};
  for (int k0 = 0; k0 < 128; k0 += 64) {
    for (int e = tid; e < 64 * 32; e += 256) {     // async-stage Wa2 slice
      int kk = e >> 5, seg = e & 31;
      async_load_b128(lds_off(&lB[kk][seg * 4]),
                      Wa2 + (size_t)(osel * 128 + k0 + kk) * 128 + seg * 4);
    }
    for (int e = tid; e < 64 * 64; e += 256) {     // stage relu(nf)
      int r = e >> 6, cc = e & 63;
      float v = nf[(mtile * 64 + r) * 128 + k0 + cc];
      lA[r][cc] = v > 0 ? v : 0;
    }
    wait_async0();
    __syncthreads();
#pragma unroll 4
    for (int kk = 0; kk < 64; kk += 4) {
      v2f a;
      a.x = lA[Mb + lane15][kk + 2 * hi];
      a.y = lA[Mb + lane15][kk + 2 * hi + 1];
#pragma unroll
      for (int t = 0; t < 4; ++t) {
        v2f b;
        b.x = lB[kk + 2 * hi][Nb + 16 * t + lane15];
        b.y = lB[kk + 2 * hi + 1][Nb + 16 * t + lane15];
        c[t] = wmma4(a, b, c[t]);
      }
    }
    __syncthreads();
  }
  for (int t = 0; t < 4; ++t)
    for (int r = 0; r < 8; ++r) {
      int m = mtile * 64 + Mb + r + 8 * hi;
      int n = Nb + 16 * t + lane15;
      out[m * 128 + n] = c[t][r];
    }
}

// ============================ Kernel 4 =====================================
// Dp[64,128] = s[64,100] @ Wa2[256:356] + ba2   (single block; K padded to 104)
__global__ void k_spec_proj(const float* __restrict__ s,
                            const float* __restrict__ Wa2,
                            const float* __restrict__ ba2,
                            float* __restrict__ Dp) {
  const int tid = threadIdx.x;
  const int w = tid >> 5, lane = tid & 31;
  const int lane15 = lane & 15, hi = lane >> 4;
  const int Mb = (w & 3) * 16, Nb = (w >> 2) * 64;
  v8f c[4] = {};
  for (int k = 0; k < 104; k += 4) {
    int ka = k + 2 * hi;
    v2f a;
    a.x = (ka < 100) ? s[(Mb + lane15) * 100 + ka] : 0.0f;
    a.y = (ka + 1 < 100) ? s[(Mb + lane15) * 100 + ka + 1] : 0.0f;
#pragma unroll
    for (int t = 0; t < 4; ++t) {
      int n = Nb + 16 * t + lane15;
      v2f b;
      b.x = (ka < 100) ? Wa2[(256 + ka) * 128 + n] : 0.0f;
      b.y = (ka + 1 < 100) ? Wa2[(257 + ka) * 128 + n] : 0.0f;
      c[t] = wmma4(a, b, c[t]);
    }
  }
  for (int t = 0; t < 4; ++t)
    for (int r = 0; r < 8; ++r) {
      int m = Mb + r + 8 * hi, n = Nb + 16 * t + lane15;
      Dp[m * 128 + n] = c[t][r] + ba2[n];
    }
}

// ============================ Kernel 5 =====================================
// Per-graph: 243 logits = relu(Ai[i]+Bj[j]+Dp[b]) @ Wf + bf, gather, softmax.
__global__ void k_actions(const float* __restrict__ Ai, const float* __restrict__ Bj,
                          const float* __restrict__ Dp, const float* __restrict__ Wf,
                          const float* __restrict__ bf, const int* __restrict__ indexmask,
                          const float* __restrict__ mask, float* __restrict__ out) {
  __shared__ float sAi[9][128], sBj[9][128], sD[128], sWf[384];
  __shared__ float fp[244];
  __shared__ float red[256];
  const int b = blockIdx.x, t = threadIdx.x;
  for (int e = t; e < 9 * 128; e += 256) {
    int i = e >> 7, cch = e & 127;
    sAi[i][cch] = Ai[(b * 9 + i) * 128 + cch];
    sBj[i][cch] = Bj[(b * 9 + i) * 128 + cch];
  }
  if (t < 128) sD[t] = Dp[b * 128 + t];
  for (int e = t; e < 384; e += 256) sWf[e] = Wf[e];
  __syncthreads();
  float val = -INFINITY;
  if (t < 243) {
    int pair = t / 3, bond = t - pair * 3;
    int i = pair / 9, j = pair - i * 9;
    float sum = bf[bond];
#pragma unroll 8
    for (int cch = 0; cch < 128; ++cch) {
      float hv = sAi[i][cch] + sBj[j][cch] + sD[cch];
      hv = hv > 0 ? hv : 0;
      sum += hv * sWf[cch * 3 + bond];
    }
    fp[t] = sum;
  }
  __syncthreads();
  if (t < 243) val = fp[indexmask[b * 243 + t]] + mask[b * 243 + t];
  red[t] = val;
  __syncthreads();
  for (int sft = 128; sft > 0; sft >>= 1) {
    if (t < sft) red[t] = fmaxf(red[t], red[t + sft]);
    __syncthreads();
  }
  float mx = red[0];
  __syncthreads();
  float ev = (t < 243) ? __expf(val - mx) : 0.0f;
  red[t] = ev;
  __syncthreads();
  for (int sft = 128; sft > 0; sft >>= 1) {
    if (t < sft) red[t] += red[t + sft];
    __syncthreads();
  }
  float sm = red[0];
  if (t < 243) out[b * 243 + t] = ev / sm;
}

// ============================ Kernel 6 =====================================
__global__ void k_readout(const float* __restrict__ nf, float* __restrict__ ro) {
  int t = blockIdx.x * blockDim.x + threadIdx.x;
  if (t < 64 * 128) {
    int b = t >> 7, cch = t & 127;
    float sum = 0;
#pragma unroll
    for (int i = 0; i < 9; ++i) sum += nf[(b * 9 + i) * 128 + cch];
    ro[t] = sum;
  }
}

// v[64] = relu([ro|s][64,228] @ Wv1[228,64] + bv1) @ Wv2[64] + bv2  (1 block)
__global__ void k_value(const float* __restrict__ ro, const float* __restrict__ s,
                        const float* __restrict__ Wv1, const float* __restrict__ bv1,
                        const float* __restrict__ Wv2, const float* __restrict__ bv2,
                        float* __restrict__ vout) {
  __shared__ float lH[64][65];
  const int tid = threadIdx.x;
  const int w = tid >> 5, lane = tid & 31;
  const int lane15 = lane & 15, hi = lane >> 4;
  const int Mb = (w & 3) * 16, Nb = (w >> 2) * 32;
  v8f c0 = {}, c1 = {};
  for (int k = 0; k < 228; k += 4) {               // 228 % 4 == 0
    int ka = k + 2 * hi, m = Mb + lane15;
    v2f a, b0, b1v;
    a.x = (ka < 128) ? ro[m * 128 + ka] : s[m * 100 + ka - 128];
    a.y = (ka + 1 < 128) ? ro[m * 128 + ka + 1] : s[m * 100 + ka + 1 - 128];
    b0.x  = Wv1[ka * 64 + Nb + lane15];
    b0.y  = Wv1[(ka + 1) * 64 + Nb + lane15];
    b1v.x = Wv1[ka * 64 + Nb + 16 + lane15];
    b1v.y = Wv1[(ka + 1) * 64 + Nb + 16 + lane15];
    c0 = wmma4(a, b0, c0);
    c1 = wmma4(a, b1v, c1);
  }
  for (int r = 0; r < 8; ++r) {
    int m = Mb + r + 8 * hi;
    float v0 = c0[r] + bv1[Nb + lane15];
    float v1 = c1[r] + bv1[Nb + 16 + lane15];
    lH[m][Nb + lane15] = v0 > 0 ? v0 : 0;
    lH[m][Nb + 16 + lane15] = v1 > 0 ? v1 : 0;
  }
  __syncthreads();
  if (tid < 64) {
    float v = bv2[0];
#pragma unroll 8
    for (int o = 0; o < 64; ++o) v += lH[tid][o] * Wv2[o];
    vout[tid] = v;
  }
}

// ===========================================================================
extern "C" void kernel_launch(void* const* d_in, const int* in_sizes, int n_in,
                              void* d_out, int out_size, void* d_ws, size_t ws_size,
                              hipStream_t stream) {
  (void)in_sizes; (void)n_in; (void)out_size; (void)ws_size;
  const float* nf        = (const float*)d_in[0];
  const float* specs     = (const float*)d_in[1];
  /* d_in[2] = len_vec: block-diagonal structure is implicit, unused */
  const float* mask      = (const float*)d_in[3];
  const int*   indexmask = (const int*)d_in[4];
  const float* W1  = (const float*)d_in[5];
  const float* b1  = (const float*)d_in[6];
  const float* W2  = (const float*)d_in[7];
  const float* b2  = (const float*)d_in[8];
  const float* Wv1 = (const float*)d_in[9];
  const float* bv1 = (const float*)d_in[10];
  const float* Wv2 = (const float*)d_in[11];
  const float* bv2 = (const float*)d_in[12];
  const float* Wa2 = (const float*)d_in[13];
  const float* ba2 = (const float*)d_in[14];
  const float* Wf  = (const float*)d_in[15];
  const float* bfv = (const float*)d_in[16];
  float* out = (float*)d_out;
  float* ws  = (float*)d_ws;

  float* s1 = ws;              // [64,900]
  float* s  = s1 + 64 * 900;   // [64,100]
  float* Ai = s  + 64 * 100;   // [576,128]
  float* Bj = Ai + 576 * 128;  // [576,128]
  float* Dp = Bj + 576 * 128;  // [64,128]
  float* ro = Dp + 64 * 128;   // [64,128]

  k_spec_gemm1<<<15, 256, 0, stream>>>(specs, W1, b1, s1);
  k_spec_gemm2<<<1, 256, 0, stream>>>(s1, W2, b2, s);
  k_pair_proj<<<dim3(9, 2), 256, 0, stream>>>(nf, Wa2, Ai, Bj);
  k_spec_proj<<<1, 256, 0, stream>>>(s, Wa2, ba2, Dp);
  k_readout<<<32, 256, 0, stream>>>(nf, ro);
  k_value<<<1, 256, 0, stream>>>(ro, s, Wv1, bv1, Wv2, bv2, out + 64 * 243);
  k_actions<<<64, 256, 0, stream>>>(Ai, Bj, Dp, Wf, bfv, indexmask, mask, out);
}